// AllSetTransformerLayer_18425409700245
// MI455X (gfx1250) — compile-verified
//
#include <hip/hip_runtime.h>
#include <hip/hip_bf16.h>

typedef __attribute__((ext_vector_type(2))) float v2f;
typedef __attribute__((ext_vector_type(8))) float v8f;

#define C_DIM 128
#define H_DIM 4
#define D_DIM 32
#define LDSP 132   // padded row pitch (floats) -> conflict-free column reads

__device__ __forceinline__ v8f wmma4(v2f a, v2f b, v8f c) {
    // D = A(16x4,f32) * B(4x16,f32) + C(16x16,f32)
    return __builtin_amdgcn_wmma_f32_16x16x4_f32(false, a, false, b, (short)0, c, false, false);
}

__device__ __forceinline__ unsigned f2ord(float f) {
    unsigned u = __float_as_uint(f);
    return (u & 0x80000000u) ? ~u : (u | 0x80000000u);
}
__device__ __forceinline__ float ord2f(unsigned u) {
    return (u & 0x80000000u) ? __uint_as_float(u & 0x7FFFFFFFu) : __uint_as_float(~u);
}

// ---------------- kq[h][c] = sum_d Kw[h,c,d] * Qw[h,0,d] ----------------
__global__ void k_kq(const float* __restrict__ Kw, const float* __restrict__ Qw,
                     float* __restrict__ kq) {
    int i = threadIdx.x;              // 512 threads: h = i>>7, c = i&127
    int h = i >> 7, c = i & 127;
    float a = 0.f;
    const float* kp = Kw + ((size_t)h * C_DIM + c) * D_DIM;
    const float* qp = Qw + (size_t)h * D_DIM;
#pragma unroll
    for (int d = 0; d < D_DIM; ++d) a += kp[d] * qp[d];
    kq[i] = a;
}

// ---------------- logit[n][h] = x[n,:] . kq[h,:]  (one wave per node) ----------------
__global__ __launch_bounds__(256) void k_logits(const float* __restrict__ x,
                                                const float* __restrict__ kq,
                                                float* __restrict__ logit, int n_src) {
    int wave = blockIdx.x * 8 + (threadIdx.x >> 5);
    int lane = threadIdx.x & 31;
    if (wave >= n_src) return;
    const float* xr = x + (size_t)wave * C_DIM;
    float a0 = 0.f, a1 = 0.f, a2 = 0.f, a3 = 0.f;
#pragma unroll
    for (int k = 0; k < 4; ++k) {
        int c = lane + 32 * k;
        float xv = xr[c];
        a0 += xv * kq[c];
        a1 += xv * kq[128 + c];
        a2 += xv * kq[256 + c];
        a3 += xv * kq[384 + c];
    }
#pragma unroll
    for (int off = 16; off; off >>= 1) {
        a0 += __shfl_xor(a0, off, 32);
        a1 += __shfl_xor(a1, off, 32);
        a2 += __shfl_xor(a2, off, 32);
        a3 += __shfl_xor(a3, off, 32);
    }
    if (lane == 0) {
        float* lp = logit + (size_t)wave * 4;
        lp[0] = a0; lp[1] = a1; lp[2] = a2; lp[3] = a3;
    }
}

// ---------------- xV[n, h*32+d] = sum_c x[n,c] * Vw[h,c,d]  (WMMA f32 16x16x4) ----------------
__global__ __launch_bounds__(256) void k_xv(const float* __restrict__ x,
                                            const float* __restrict__ Vw,
                                            float* __restrict__ xV, int n_src) {
    __shared__ float xs[16][LDSP];
    const int rowbase = blockIdx.x * 16;
    const int tid = threadIdx.x;
    for (int idx = tid; idx < 16 * 128; idx += 256) {
        int r = idx >> 7, c = idx & 127;
        int g = rowbase + r;
        xs[r][c] = (g < n_src) ? x[(size_t)g * C_DIM + c] : 0.f;
    }
    __syncthreads();
    const int wave = tid >> 5, lane = tid & 31;
    const int half = lane >> 4, l16 = lane & 15;
    const int colbase = wave * 16;              // 16-col tile lies within one head
    const int h = colbase >> 5;
    const int d = (colbase & 31) + l16;
    const float* Bp = Vw + (size_t)h * C_DIM * D_DIM + d;   // Vw[(h*128+k)*32 + d]
    v8f acc = {0.f, 0.f, 0.f, 0.f, 0.f, 0.f, 0.f, 0.f};
#pragma unroll
    for (int k = 0; k < 128; k += 4) {
        int kk = k + half * 2;
        v2f a; a.x = xs[l16][kk];       a.y = xs[l16][kk + 1];
        v2f b; b.x = Bp[(size_t)kk * 32]; b.y = Bp[(size_t)(kk + 1) * 32];
        acc = wmma4(a, b, acc);
    }
#pragma unroll
    for (int r = 0; r < 8; ++r) {
        int g = rowbase + r + half * 8;
        if (g < n_src) xV[(size_t)g * C_DIM + colbase + l16] = acc[r];
    }
}

// ---------------- init helpers ----------------
__global__ void k_fill_f32(float* p, int n, float v) {
    int i = blockIdx.x * 256 + threadIdx.x;
    if (i < n) p[i] = v;
}
__global__ void k_fill_u32(unsigned* p, int n, unsigned v) {
    int i = blockIdx.x * 256 + threadIdx.x;
    if (i < n) p[i] = v;
}

// ---------------- segment max over edges (ordered-uint atomicMax) ----------------
__global__ void k_segmax(const int* __restrict__ src, const int* __restrict__ tgt,
                         const float* __restrict__ logit, unsigned* __restrict__ menc, int E) {
    int e = blockIdx.x * 256 + threadIdx.x;
    if (e >= E) return;
    int n = src[e], t = tgt[e];
#pragma unroll
    for (int h = 0; h < H_DIM; ++h)
        atomicMax(menc + (size_t)t * 4 + h, f2ord(logit[(size_t)n * 4 + h]));
}

// ---------------- edge scatter (one wave32 per edge, float4 lanes) ----------------
// ex = exp(l - m); s[t,h] += ex; msg[t,:] += ex * xV[src,:]
__global__ __launch_bounds__(256) void k_scatter(const int* __restrict__ src,
                                                 const int* __restrict__ tgt,
                                                 const float* __restrict__ logit,
                                                 const unsigned* __restrict__ menc,
                                                 const float* __restrict__ xV,
                                                 float* __restrict__ ssum,
                                                 float* __restrict__ msg, int E) {
    int e = blockIdx.x * 8 + (threadIdx.x >> 5);   // wave per edge (uniform exit)
    if (e >= E) return;
    const int lane = threadIdx.x & 31;
    const int n = src[e], t = tgt[e];              // uniform-address loads (1 request)
    const int h = lane >> 3;                       // lane's 4 columns lie in one head
    float mx = ord2f(menc[(size_t)t * 4 + h]);
    float ex = __expf(logit[(size_t)n * 4 + h] - mx);
    if ((lane & 7) == 0) atomicAdd(ssum + (size_t)t * 4 + h, ex);  // lanes 0,8,16,24
    const float4 xv = *(const float4*)(xV + (size_t)n * 128 + lane * 4);  // b128 gather
    float* mp = msg + (size_t)t * 128 + lane * 4;
    atomicAdd(mp + 0, ex * xv.x);
    atomicAdd(mp + 1, ex * xv.y);
    atomicAdd(mp + 2, ex * xv.z);
    atomicAdd(mp + 3, ex * xv.w);
}

// ---------------- fused tail: bias + LN0 + (W1,relu) + (W2,relu) + residual + LN1 + relu ----------------
__global__ __launch_bounds__(256) void k_tail(const float* __restrict__ msg,
                                              const float* __restrict__ ssum,
                                              const float* __restrict__ Qw,
                                              const float* __restrict__ g0, const float* __restrict__ b0,
                                              const float* __restrict__ W1, const float* __restrict__ W2,
                                              const float* __restrict__ g1, const float* __restrict__ b1,
                                              float* __restrict__ out, int n_tgt) {
    __shared__ float hln[16][LDSP];
    __shared__ float bufA[16][LDSP];
    __shared__ float bufB[16][LDSP];
    const int rowbase = blockIdx.x * 16;
    const int tid = threadIdx.x;
    const int wave = tid >> 5, lane = tid & 31;
    const int half = lane >> 4, l16 = lane & 15;

    // 1) raw h = msg/s + Qw bias  -> bufA
    for (int idx = tid; idx < 16 * 128; idx += 256) {
        int r = idx >> 7, c = idx & 127;
        int t = rowbase + r;
        float v = 0.f;
        if (t < n_tgt) {
            float sv = ssum[(size_t)t * 4 + (c >> 5)];
            float mv = msg[(size_t)t * 128 + c];
            v = (sv > 0.f ? mv / sv : 0.f) + Qw[c];
        }
        bufA[r][c] = v;
    }
    __syncthreads();

    // 2) LN0 per row (2 rows per wave) -> hln
#pragma unroll
    for (int rr = 0; rr < 2; ++rr) {
        int r = wave * 2 + rr;
        float v0 = bufA[r][lane], v1 = bufA[r][lane + 32];
        float v2 = bufA[r][lane + 64], v3 = bufA[r][lane + 96];
        float s = v0 + v1 + v2 + v3;
        float q = v0 * v0 + v1 * v1 + v2 * v2 + v3 * v3;
#pragma unroll
        for (int off = 16; off; off >>= 1) {
            s += __shfl_xor(s, off, 32);
            q += __shfl_xor(q, off, 32);
        }
        float mean = s * (1.f / 128.f);
        float var = q * (1.f / 128.f) - mean * mean;
        float rstd = rsqrtf(var + 1e-5f);
        hln[r][lane]      = (v0 - mean) * rstd * g0[lane]      + b0[lane];
        hln[r][lane + 32] = (v1 - mean) * rstd * g0[lane + 32] + b0[lane + 32];
        hln[r][lane + 64] = (v2 - mean) * rstd * g0[lane + 64] + b0[lane + 64];
        hln[r][lane + 96] = (v3 - mean) * rstd * g0[lane + 96] + b0[lane + 96];
    }
    __syncthreads();

    const int colbase = wave * 16;
    const int col = colbase + l16;

    // 3) m1 = relu(hln @ W1) -> bufA
    {
        v8f acc = {0.f, 0.f, 0.f, 0.f, 0.f, 0.f, 0.f, 0.f};
#pragma unroll
        for (int k = 0; k < 128; k += 4) {
            int kk = k + half * 2;
            v2f a; a.x = hln[l16][kk];               a.y = hln[l16][kk + 1];
            v2f b; b.x = W1[(size_t)kk * 128 + col]; b.y = W1[(size_t)(kk + 1) * 128 + col];
            acc = wmma4(a, b, acc);
        }
#pragma unroll
        for (int r = 0; r < 8; ++r) bufA[r + half * 8][col] = fmaxf(acc[r], 0.f);
    }
    __syncthreads();

    // 4) m2 = relu(m1 @ W2) -> bufB
    {
        v8f acc = {0.f, 0.f, 0.f, 0.f, 0.f, 0.f, 0.f, 0.f};
#pragma unroll
        for (int k = 0; k < 128; k += 4) {
            int kk = k + half * 2;
            v2f a; a.x = bufA[l16][kk];              a.y = bufA[l16][kk + 1];
            v2f b; b.x = W2[(size_t)kk * 128 + col]; b.y = W2[(size_t)(kk + 1) * 128 + col];
            acc = wmma4(a, b, acc);
        }
#pragma unroll
        for (int r = 0; r < 8; ++r) bufB[r + half * 8][col] = fmaxf(acc[r], 0.f);
    }
    __syncthreads();

    // 5) out = relu(LN1(hln + m2))
#pragma unroll
    for (int rr = 0; rr < 2; ++rr) {
        int r = wave * 2 + rr;
        int t = rowbase + r;
        float v0 = hln[r][lane]      + bufB[r][lane];
        float v1 = hln[r][lane + 32] + bufB[r][lane + 32];
        float v2 = hln[r][lane + 64] + bufB[r][lane + 64];
        float v3 = hln[r][lane + 96] + bufB[r][lane + 96];
        float s = v0 + v1 + v2 + v3;
        float q = v0 * v0 + v1 * v1 + v2 * v2 + v3 * v3;
#pragma unroll
        for (int off = 16; off; off >>= 1) {
            s += __shfl_xor(s, off, 32);
            q += __shfl_xor(q, off, 32);
        }
        float mean = s * (1.f / 128.f);
        float var = q * (1.f / 128.f) - mean * mean;
        float rstd = rsqrtf(var + 1e-5f);
        if (t < n_tgt) {
            float* op = out + (size_t)t * 128;
            op[lane]      = fmaxf((v0 - mean) * rstd * g1[lane]      + b1[lane], 0.f);
            op[lane + 32] = fmaxf((v1 - mean) * rstd * g1[lane + 32] + b1[lane + 32], 0.f);
            op[lane + 64] = fmaxf((v2 - mean) * rstd * g1[lane + 64] + b1[lane + 64], 0.f);
            op[lane + 96] = fmaxf((v3 - mean) * rstd * g1[lane + 96] + b1[lane + 96], 0.f);
        }
    }
}

// ---------------- host-side pass driver ----------------
struct Ws {
    float* kq; float* logit; unsigned* menc; float* ssum; float* xV; float* msg;
};

static void run_block(const float* x, const int* tgt, const int* src,
                      int n_src, int n_tgt, int E,
                      const float* Kw, const float* Qw, const float* Vw,
                      const float* g0, const float* b0,
                      const float* W1, const float* W2,
                      const float* g1, const float* b1,
                      float* out, const Ws& w, hipStream_t stream) {
    k_kq<<<1, 512, 0, stream>>>(Kw, Qw, w.kq);
    k_logits<<<(n_src + 7) / 8, 256, 0, stream>>>(x, w.kq, w.logit, n_src);
    k_xv<<<(n_src + 15) / 16, 256, 0, stream>>>(x, Vw, w.xV, n_src);
    k_fill_u32<<<(n_tgt * 4 + 255) / 256, 256, 0, stream>>>(w.menc, n_tgt * 4, 0x007FFFFFu); // enc(-inf)
    k_fill_f32<<<(n_tgt * 4 + 255) / 256, 256, 0, stream>>>(w.ssum, n_tgt * 4, 0.f);
    k_fill_f32<<<(n_tgt * 128 + 255) / 256, 256, 0, stream>>>(w.msg, n_tgt * 128, 0.f);
    k_segmax<<<(E + 255) / 256, 256, 0, stream>>>(src, tgt, w.logit, w.menc, E);
    k_scatter<<<(E + 7) / 8, 256, 0, stream>>>(src, tgt, w.logit, w.menc,
                                               w.xV, w.ssum, w.msg, E);
    k_tail<<<(n_tgt + 15) / 16, 256, 0, stream>>>(w.msg, w.ssum, Qw, g0, b0, W1, W2, g1, b1,
                                                  out, n_tgt);
}

extern "C" void kernel_launch(void* const* d_in, const int* in_sizes, int n_in,
                              void* d_out, int out_size, void* d_ws, size_t ws_size,
                              hipStream_t stream) {
    const float* x0        = (const float*)d_in[0];
    const int*   node_idx  = (const int*)d_in[1];   // sources for v2e, targets for e2v
    const int*   hedge_idx = (const int*)d_in[2];   // targets for v2e, sources for e2v
    const int N = in_sizes[0] / C_DIM;
    const int E = in_sizes[1];
    const int M = out_size / C_DIM - N;

    const float* v2e_K = (const float*)d_in[5];
    const float* v2e_Q = (const float*)d_in[6];
    const float* v2e_V = (const float*)d_in[7];
    const float* v2e_g0 = (const float*)d_in[8];
    const float* v2e_b0 = (const float*)d_in[9];
    const float* v2e_W1 = (const float*)d_in[10];
    const float* v2e_W2 = (const float*)d_in[11];
    const float* v2e_g1 = (const float*)d_in[12];
    const float* v2e_b1 = (const float*)d_in[13];
    const float* e2v_K = (const float*)d_in[14];
    const float* e2v_Q = (const float*)d_in[15];
    const float* e2v_V = (const float*)d_in[16];
    const float* e2v_g0 = (const float*)d_in[17];
    const float* e2v_b0 = (const float*)d_in[18];
    const float* e2v_W1 = (const float*)d_in[19];
    const float* e2v_W2 = (const float*)d_in[20];
    const float* e2v_g1 = (const float*)d_in[21];
    const float* e2v_b1 = (const float*)d_in[22];

    float* x0_out = (float*)d_out;                      // [N,128]
    float* x1     = (float*)d_out + (size_t)N * C_DIM;  // [M,128]

    // carve workspace (sized for the larger of the two passes)
    const int maxNT = (N > M) ? N : M;
    char* p = (char*)d_ws;
    auto take = [&](size_t bytes) { char* r = p; p += (bytes + 255) & ~(size_t)255; return r; };
    Ws w;
    w.kq    = (float*)take(512 * sizeof(float));
    w.logit = (float*)take((size_t)maxNT * 4 * sizeof(float));
    w.menc  = (unsigned*)take((size_t)maxNT * 4 * sizeof(unsigned));
    w.ssum  = (float*)take((size_t)maxNT * 4 * sizeof(float));
    w.xV    = (float*)take((size_t)maxNT * C_DIM * sizeof(float));
    w.msg   = (float*)take((size_t)maxNT * C_DIM * sizeof(float));

    // pass 1: vertex -> hyperedge (targets = hyperedges)
    run_block(x0, hedge_idx, node_idx, N, M, E,
              v2e_K, v2e_Q, v2e_V, v2e_g0, v2e_b0, v2e_W1, v2e_W2, v2e_g1, v2e_b1,
              x1, w, stream);
    // pass 2: hyperedge -> vertex (targets = nodes)
    run_block(x1, node_idx, hedge_idx, M, N, E,
              e2v_K, e2v_Q, e2v_V, e2v_g0, e2v_b0, e2v_W1, e2v_W2, e2v_g1, e2v_b1,
              x0_out, w, stream);
}